// LangExtractor_10385230921831
// MI455X (gfx1250) — compile-verified
//
#include <hip/hip_runtime.h>

typedef float v2f __attribute__((ext_vector_type(2)));
typedef float v4f __attribute__((ext_vector_type(4)));
typedef float v8f __attribute__((ext_vector_type(8)));

#define B_TOTAL 131072
#define N_OBJ   5
#define D_F     12
#define E_F     64
#define H_F     32
#define BPG     16                       // batches per workgroup
#define ROWS    (BPG * N_OBJ)            // 80 obs rows per workgroup
#define REC     (D_F + E_F + N_OBJ)      // 81 floats per object record
#define OUT_PB  (N_OBJ * REC)            // 405 floats per batch
#define UVP     (E_F + 1)                // padded LDS row stride (bank-conflict free)

__global__ __launch_bounds__(256)
void relnet_fused(const float* __restrict__ obs,
                  const float* __restrict__ ghat,
                  const float* __restrict__ W1,
                  const float* __restrict__ b1,
                  const float* __restrict__ W2,
                  const float* __restrict__ b2,
                  float* __restrict__ out)
{
    __shared__ float s_uv[ROWS][UVP];    // u in cols [0,32), v in cols [32,64)
    __shared__ float s_g2[BPG][H_F];     // (W2 @ ghat_b)[h] per batch
    __shared__ float s_b2g[BPG];         // b2 . ghat_b per batch
    __shared__ float s_zhat[BPG][N_OBJ];
    __shared__ float s_b1[H_F];
    __shared__ float s_c2[H_F];          // W2 @ ones_E
    __shared__ float s_sb2;              // sum(b2)

    const int tid   = threadIdx.x;
    const int lane  = tid & 31;
    const int wave  = tid >> 5;
    const int lo    = lane & 15;
    const int hi    = lane >> 4;
    const int klo   = 2 * hi;            // A/B fragment K offset per ISA layout
    const int bbase = blockIdx.x * BPG;

    // ---------------- Phase 1: WMMA GEMMs + aux (wave-uniform branches) -----
    if (wave < 5) {
        // UV GEMM: [80 x 12] obs-rows @ [12 x 64] (W1_top | W1_bot)
        const int  lr   = wave * 16 + lo;                 // local row 0..79
        const long grow = (long)bbase * N_OBJ + lr;       // global obs row
        const float* arow = obs + grow * D_F;
        #pragma unroll
        for (int nt = 0; nt < 4; ++nt) {
            const int wbase = (nt < 2) ? 0 : D_F;         // top vs bottom half of W1
            const int wcol  = ((nt & 1) * 16) + lo;       // W1 column (0..31)
            v8f c = {};
            #pragma unroll
            for (int k0 = 0; k0 < D_F; k0 += 4) {
                v2f a, b;
                a.x = arow[k0 + klo];
                a.y = arow[k0 + klo + 1];
                b.x = W1[(wbase + k0 + klo    ) * H_F + wcol];
                b.y = W1[(wbase + k0 + klo + 1) * H_F + wcol];
                c = __builtin_amdgcn_wmma_f32_16x16x4_f32(
                        false, a, false, b, (short)0, c, false, false);
            }
            #pragma unroll
            for (int r = 0; r < 8; ++r)
                s_uv[wave * 16 + r + 8 * hi][nt * 16 + lo] = c[r];
        }
    } else if (wave < 7) {
        // g2 GEMM: ghat[16 x 64] @ W2^T[64 x 32]
        const int n0   = (wave - 5) * 16;
        const int hcol = n0 + lo;                         // hidden channel 0..31
        v8f c = {};
        for (int k0 = 0; k0 < E_F; k0 += 4) {
            v2f a, b;
            const float* g = ghat + (long)(bbase + lo) * E_F;
            a.x = g[k0 + klo];
            a.y = g[k0 + klo + 1];
            b.x = W2[hcol * E_F + k0 + klo];
            b.y = W2[hcol * E_F + k0 + klo + 1];
            c = __builtin_amdgcn_wmma_f32_16x16x4_f32(
                    false, a, false, b, (short)0, c, false, false);
        }
        #pragma unroll
        for (int r = 0; r < 8; ++r)
            s_g2[r + 8 * hi][hcol] = c[r];
    } else {
        // wave 7: aux scalars/vectors
        const float b2lo = b2[lane], b2hi = b2[lane + 32];
        s_b1[lane] = b1[lane];
        float cs = 0.f;
        for (int e = 0; e < E_F; ++e) cs += W2[lane * E_F + e];
        s_c2[lane] = cs;
        float sb = b2lo + b2hi;
        #pragma unroll
        for (int m = 16; m >= 1; m >>= 1) sb += __shfl_xor(sb, m, 32);
        if (lane == 0) s_sb2 = sb;
        for (int bl = 0; bl < BPG; ++bl) {
            const float* g = ghat + (long)(bbase + bl) * E_F;
            float t = b2lo * g[lane] + b2hi * g[lane + 32];
            #pragma unroll
            for (int m = 16; m >= 1; m >>= 1) t += __shfl_xor(t, m, 32);
            if (lane == 0) s_b2g[bl] = t;
        }
    }
    __syncthreads();

    // ---------------- Phase 2: per-batch attention (2 batches per wave) -----
    #pragma unroll
    for (int bb = 0; bb < 2; ++bb) {
        const int bl = wave * 2 + bb;
        const int j  = (lane < 25) ? (lane / 5) : 4;   // clamp inactive lanes
        const int k  = (lane < 25) ? (lane % 5) : 4;
        const float* urow = &s_uv[bl * N_OBJ + j][0];
        const float* vrow = &s_uv[bl * N_OBJ + k][H_F];
        float aw = 0.f, az = 0.f;
        #pragma unroll
        for (int h = 0; h < H_F; ++h) {
            float hid = urow[h] + vrow[h] + s_b1[h];
            hid = fmaxf(hid, 0.f);
            aw  = fmaf(hid, s_g2[bl][h], aw);
            az  = fmaf(hid, s_c2[h],    az);
        }
        const float w  = (lane < 25) ? (aw + s_b2g[bl]) : -__builtin_inff();
        const float zs = az + s_sb2;
        float wm = w;
        #pragma unroll
        for (int m = 16; m >= 1; m >>= 1) wm = fmaxf(wm, __shfl_xor(wm, m, 32));
        const float p = __expf(w - wm);               // 0 for inactive lanes
        float ps = p;
        #pragma unroll
        for (int m = 16; m >= 1; m >>= 1) ps += __shfl_xor(ps, m, 32);
        const float q = p * zs / ps;
        float zh = 0.f;                               // lane j<5 gathers its 5 k's
        #pragma unroll
        for (int d = 0; d < 5; ++d) zh += __shfl(q, lane * 5 + d, 32);
        if (lane < N_OBJ) s_zhat[bl][lane] = zh;
    }
    __syncthreads();

    // ------- Phase 3: streaming output, 16B-aligned non-temporal b128 -------
    // WG output block = 16*405 floats = 1620 aligned float4 chunks.
    const long obase = (long)bbase * OUT_PB;
    for (int c = tid; c < (BPG * OUT_PB) / 4; c += 256) {
        v4f v;
        #pragma unroll
        for (int e = 0; e < 4; ++e) {
            const int idx = c * 4 + e;
            const int bl  = idx / OUT_PB;
            const int r4  = idx % OUT_PB;
            const int j   = r4 / REC;
            const int r   = r4 % REC;
            float val;
            if (r < D_F)
                val = obs[((long)(bbase + bl) * N_OBJ + j) * D_F + r];
            else if (r < D_F + E_F)
                val = ghat[(long)(bbase + bl) * E_F + (r - D_F)];
            else
                val = s_zhat[bl][r - (D_F + E_F)];
            v[e] = val;
        }
        __builtin_nontemporal_store(v, (v4f*)(out + obase + (long)c * 4));
    }
}

extern "C" void kernel_launch(void* const* d_in, const int* in_sizes, int n_in,
                              void* d_out, int out_size, void* d_ws, size_t ws_size,
                              hipStream_t stream) {
    const float* obs  = (const float*)d_in[0];
    const float* ghat = (const float*)d_in[1];
    const float* W1   = (const float*)d_in[2];
    const float* b1   = (const float*)d_in[3];
    const float* W2   = (const float*)d_in[4];
    const float* b2   = (const float*)d_in[5];
    float* out = (float*)d_out;
    relnet_fused<<<dim3(B_TOTAL / BPG), 256, 0, stream>>>(obs, ghat, W1, b1, W2, b2, out);
}